// FFTSplitAdaptive_20220706029719
// MI455X (gfx1250) — compile-verified
//
#include <hip/hip_runtime.h>
#include <math.h>

// ---------------------------------------------------------------------------
// FFTSplitAdaptive for MI455X (gfx1250, wave32).
//  x: (16,3,512,512) f32.  Outputs concatenated:
//  i_high[48*512*512], i_low[48*512*512], d0[16], mask_low[16*512*512],
//  mask_high[16*512*512]   (all float32)
//
//  i_high = x - i_low  (mask_high = 1 - mask_low)  => only ONE inverse FFT.
//  512-pt FFT per wave32, four-step 512 = 16 x 32:
//    16 x (radix-2 Stockham FFT-32 in LDS) -> W512 twiddle ->
//    DFT-16 over 32 columns as complex 16x16x32 GEMM on
//    V_WMMA_F32_16X16X4_F32 (4 chained K-chunks, 4 real matmuls per half).
//  All twiddles come from one 512-entry LDS table (built once per block);
//  inverse transforms conjugate on read.
// ---------------------------------------------------------------------------

typedef __attribute__((ext_vector_type(2))) float v2f;
typedef __attribute__((ext_vector_type(8))) float v8f;

#define WMMA4(A, B, C) \
    __builtin_amdgcn_wmma_f32_16x16x4_f32(false, (A), false, (B), (short)0, (C), false, false)

__device__ __forceinline__ float2 cmulf2(float2 a, float2 b) {
    return make_float2(a.x * b.x - a.y * b.y, a.x * b.y + a.y * b.x);
}

// Radial grid value, replicating the reference's float math.
__device__ __forceinline__ float rr_of(float dy, float dx) {
    const float md = 362.0386729675124f;  // sqrt(256^2+256^2)+1e-6 (f32-rounded)
    float yy = dy / md, xx = dx / md;
    return sqrtf(yy * yy + xx * xx);
}
__device__ __forceinline__ float rmax_val() { return rr_of(-256.f, -256.f); }

// Twiddle table: tw[t] = exp(-2*pi*i*t/512).  2 sincos per thread, once/block.
__device__ __forceinline__ void build_tw(float2* tw, int tid) {
    const float TWO_PI = 6.2831853071795864769f;
    #pragma unroll
    for (int k = 0; k < 2; ++k) {
        int t = tid + 256 * k;
        float sn, cs;
        __sincosf(-TWO_PI * (float)t * (1.0f / 512.0f), &sn, &cs);
        tw[t] = make_float2(cs, sn);
    }
}

// ---------------------------------------------------------------------------
// One 512-point FFT per wave32.
// Input:  bA[(i&15)*32 + (i>>4)] = x[i]  (n1 = i&15, n2 = i>>4)
// Output: bA[u] = X[u], natural order.
// im = +1 forward, -1 inverse (conjugates the twiddle table; unscaled).
// Contains __syncthreads(); all waves of the block must call uniformly.
// ---------------------------------------------------------------------------
__device__ __forceinline__ void fft512_wave(float2* bA, float2* bB,
                                            const float2* tw, int lane, float im)
{
    // ---- 16 sub-FFTs of length 32, Stockham radix-2 (autosort), 5 stages ---
    float2* src = bA;
    float2* dst = bB;
    int s = 1, ls = 0, tstep = 16;              // tstep = 512/n
    #pragma unroll
    for (int n = 32; n > 1; n >>= 1) {
        int m = n >> 1;
        #pragma unroll
        for (int t = 0; t < 8; ++t) {           // 256 butterflies / 32 lanes
            int task = lane + 32 * t;
            int f = task >> 4;                  // sub-FFT id = n1
            int b = task & 15;
            int p = b >> ls;
            int q = b & (s - 1);
            float2 a = src[f * 32 + q + s * p];
            float2 c = src[f * 32 + q + s * (p + m)];
            float2 su = make_float2(a.x + c.x, a.y + c.y);
            float2 di = make_float2(a.x - c.x, a.y - c.y);
            float2 wv = tw[tstep * p];          // LDS broadcast (shared p)
            dst[f * 32 + q + s * (2 * p)]     = su;
            dst[f * 32 + q + s * (2 * p + 1)] = cmulf2(di, make_float2(wv.x, im * wv.y));
        }
        __syncthreads();
        float2* tp = src; src = dst; dst = tp;
        s <<= 1; ++ls; tstep <<= 1;
    }
    // src == bB now holds G[n1][k2] at [n1*32 + k2].

    // ---- twiddle: Y[n1][k2] = G * W512^{+/- n1*k2} --------------------------
    #pragma unroll
    for (int t = 0; t < 16; ++t) {
        int idx = lane + 32 * t;
        int n1 = idx >> 5, k2 = idx & 31;
        float2 wv = tw[(n1 * k2) & 511];
        src[idx] = cmulf2(src[idx], make_float2(wv.x, im * wv.y));
    }
    __syncthreads();

    // ---- DFT-16 over n1 for 32 columns via WMMA f32 16x16x4 ----------------
    // Z[k1][k2] = sum_n1 W16^{k1*n1} * Y[n1][k2];  output index u = 32*k1 + k2.
    int mrow = lane & 15;
    int hi   = (lane >> 4) & 1;
    v2f Ar[4], Aip[4], Ain[4];      // Re(F16), Im(F16), -Im(F16) in A layout
    #pragma unroll
    for (int j = 0; j < 4; ++j) {
        #pragma unroll
        for (int v = 0; v < 2; ++v) {
            int kk = 4 * j + v + 2 * hi;        // documented 16x4 A layout
            float2 wv = tw[((mrow * kk) & 15) << 5];
            float ai = im * wv.y;
            Ar[j][v] = wv.x; Aip[j][v] = ai; Ain[j][v] = -ai;
        }
    }
    #pragma unroll
    for (int h = 0; h < 2; ++h) {               // two 16-column halves of N=32
        v8f zr = {0.f,0.f,0.f,0.f,0.f,0.f,0.f,0.f};
        v8f zi = {0.f,0.f,0.f,0.f,0.f,0.f,0.f,0.f};
        #pragma unroll
        for (int j = 0; j < 4; ++j) {           // K chunks of 4
            v2f Brv, Biv;
            #pragma unroll
            for (int v = 0; v < 2; ++v) {
                int row = 4 * j + v + 2 * hi;   // B 4x16 layout (rows K split 0/1 vs 2/3)
                float2 e = src[row * 32 + 16 * h + mrow];
                Brv[v] = e.x; Biv[v] = e.y;
            }
            zr = WMMA4(Ar[j],  Brv, zr);        // + Ar*Br
            zr = WMMA4(Ain[j], Biv, zr);        // - Ai*Bi
            zi = WMMA4(Ar[j],  Biv, zi);        // + Ar*Bi
            zi = WMMA4(Aip[j], Brv, zi);        // + Ai*Br
        }
        #pragma unroll
        for (int v = 0; v < 8; ++v) {           // documented 16x16 C/D layout
            int k1 = v + 8 * hi;
            int u  = 32 * k1 + 16 * h + mrow;
            bA[u] = make_float2(zr[v], zi[v]);
        }
    }
    __syncthreads();
}

// ---------------------------------------------------------------------------
// Kernels
// ---------------------------------------------------------------------------

__global__ void k_zero(float* __restrict__ p, int n) {
    int i = blockIdx.x * 256 + threadIdx.x;
    if (i < n) p[i] = 0.f;
}

// Forward FFT along rows; x real -> A complex.  8 rows / block (1 per wave).
__global__ __launch_bounds__(256)
void k_row_fwd(const float* __restrict__ x, float2* __restrict__ A)
{
    __shared__ float2 lds[8 * 1024];
    __shared__ float2 tw[512];
    int tid = threadIdx.x, lane = tid & 31, w = tid >> 5;
    int base_row = blockIdx.x * 8;
    build_tw(tw, tid);
    const float4* x4 = (const float4*)(x + (size_t)base_row * 512);
    #pragma unroll
    for (int k = 0; k < 4; ++k) {
        int idx = tid + 256 * k;                // 0..1023 float4's
        int r  = idx >> 7;
        int i0 = (idx & 127) << 2;
        float4 v = x4[idx];
        lds[r * 1024 + ((i0 + 0) & 15) * 32 + ((i0 + 0) >> 4)] = make_float2(v.x, 0.f);
        lds[r * 1024 + ((i0 + 1) & 15) * 32 + ((i0 + 1) >> 4)] = make_float2(v.y, 0.f);
        lds[r * 1024 + ((i0 + 2) & 15) * 32 + ((i0 + 2) >> 4)] = make_float2(v.z, 0.f);
        lds[r * 1024 + ((i0 + 3) & 15) * 32 + ((i0 + 3) >> 4)] = make_float2(v.w, 0.f);
    }
    __syncthreads();
    fft512_wave(&lds[w * 1024], &lds[w * 1024 + 512], tw, lane, +1.0f);
    float4* A4 = (float4*)(A + (size_t)base_row * 512);
    #pragma unroll
    for (int k = 0; k < 8; ++k) {
        int idx = tid + 256 * k;                // 0..2047 float4's (2 complex)
        int r  = idx >> 8;
        int u0 = (idx & 255) << 1;
        float2 z0 = lds[r * 1024 + u0];
        float2 z1 = lds[r * 1024 + u0 + 1];
        A4[idx] = make_float4(z0.x, z0.y, z1.x, z1.y);
    }
}

// Forward FFT along columns (in place) + radial power binning.
__global__ __launch_bounds__(256)
void k_col_fwd(float2* __restrict__ A, float* __restrict__ gbins)
{
    __shared__ float2 lds[8 * 1024];
    __shared__ float2 tw[512];
    __shared__ float binsL[101];
    int tid = threadIdx.x, lane = tid & 31, w = tid >> 5;
    int img = blockIdx.x >> 6;
    int c0  = (blockIdx.x & 63) * 8;
    size_t base = (size_t)img * 262144;
    build_tw(tw, tid);
    if (tid < 101) binsL[tid] = 0.f;
    #pragma unroll
    for (int k = 0; k < 8; ++k) {
        int idx = tid + 256 * k;                // 0..2047 float4's (2 complex)
        int n  = idx >> 2;
        int cp = (idx & 3) << 1;
        float4 e = *(const float4*)&A[base + (size_t)n * 512 + c0 + cp];
        lds[(cp + 0) * 1024 + (n & 15) * 32 + (n >> 4)] = make_float2(e.x, e.y);
        lds[(cp + 1) * 1024 + (n & 15) * 32 + (n >> 4)] = make_float2(e.z, e.w);
    }
    __syncthreads();
    fft512_wave(&lds[w * 1024], &lds[w * 1024 + 512], tw, lane, +1.0f);
    const float rscale = 100.0f / rmax_val();
    #pragma unroll
    for (int k = 0; k < 8; ++k) {
        int idx = tid + 256 * k;
        int u  = idx >> 2;
        int cp = (idx & 3) << 1;
        float2 z0 = lds[(cp + 0) * 1024 + u];
        float2 z1 = lds[(cp + 1) * 1024 + u];
        *(float4*)&A[base + (size_t)u * 512 + c0 + cp] = make_float4(z0.x, z0.y, z1.x, z1.y);
        int su = (u < 256) ? u : u - 512;       // fftshifted radial coordinate
        #pragma unroll
        for (int d = 0; d < 2; ++d) {
            int c  = c0 + cp + d;
            int sc = (c < 256) ? c : c - 512;
            float rr = rr_of((float)su, (float)sc);
            int bin = (int)floorf(rr * rscale);
            bin = min(max(bin, 0), 100);
            float2 z = d ? z1 : z0;
            atomicAdd(&binsL[bin], z.x * z.x + z.y * z.y);
        }
    }
    __syncthreads();
    if (tid < 101) atomicAdd(&gbins[(img / 3) * 101 + tid], binsL[tid]);
}

// d0 per batch from cumulative bin energy.
__global__ void k_d0(const float* __restrict__ gbins, float* __restrict__ d0ws,
                     float* __restrict__ outd0)
{
    int b = threadIdx.x;
    if (b >= 16) return;
    const float* gb = gbins + b * 101;
    float total = 0.f;
    for (int i = 0; i <= 100; ++i) total += gb[i];
    total = fmaxf(total, 1e-12f);
    float cum = 0.f; int idx = -1;
    for (int i = 0; i < 100; ++i) {
        cum += gb[i];
        if (idx < 0 && (cum / total) >= 0.95f) idx = i;
    }
    float fi = (idx < 0) ? 99.5f : ((float)idx + 0.5f);
    float d0 = fi * (rmax_val() / 100.0f);
    d0ws[b] = d0;
    outd0[b] = d0;
}

// mask_low / mask_high outputs (shifted-grid layout), float4 stores.
__global__ __launch_bounds__(256)
void k_mask(const float* __restrict__ d0ws, float* __restrict__ mlow,
            float* __restrict__ mhigh)
{
    int idx4 = blockIdx.x * 256 + threadIdx.x;  // 16*262144/4 total
    int e0  = idx4 << 2;
    int b   = e0 >> 18;
    int rem = e0 & 262143;
    int py = rem >> 9, px = rem & 511;          // px..px+3 share py
    float d0 = d0ws[b];
    float4 ml, mh;
    float* pml = (float*)&ml;
    float* pmh = (float*)&mh;
    #pragma unroll
    for (int d = 0; d < 4; ++d) {
        float rr = rr_of((float)(py - 256), (float)(px + d - 256));
        float m = (rr <= d0) ? 1.f : 0.f;
        pml[d] = m; pmh[d] = 1.f - m;
    }
    ((float4*)mlow)[idx4]  = ml;
    ((float4*)mhigh)[idx4] = mh;
}

// Inverse FFT along columns of (X * mask_low), in place (unscaled).
__global__ __launch_bounds__(256)
void k_col_inv(float2* __restrict__ A, const float* __restrict__ d0ws)
{
    __shared__ float2 lds[8 * 1024];
    __shared__ float2 tw[512];
    int tid = threadIdx.x, lane = tid & 31, w = tid >> 5;
    int img = blockIdx.x >> 6;
    int c0  = (blockIdx.x & 63) * 8;
    size_t base = (size_t)img * 262144;
    float d0 = d0ws[img / 3];
    build_tw(tw, tid);
    #pragma unroll
    for (int k = 0; k < 8; ++k) {
        int idx = tid + 256 * k;
        int n  = idx >> 2;
        int cp = (idx & 3) << 1;
        float4 e = *(const float4*)&A[base + (size_t)n * 512 + c0 + cp];
        int sn_ = (n < 256) ? n : n - 512;
        #pragma unroll
        for (int d = 0; d < 2; ++d) {
            int c  = c0 + cp + d;
            int sc = (c < 256) ? c : c - 512;
            float keep = (rr_of((float)sn_, (float)sc) <= d0) ? 1.f : 0.f;
            float2 z = d ? make_float2(e.z, e.w) : make_float2(e.x, e.y);
            lds[(cp + d) * 1024 + (n & 15) * 32 + (n >> 4)] =
                make_float2(z.x * keep, z.y * keep);
        }
    }
    __syncthreads();
    fft512_wave(&lds[w * 1024], &lds[w * 1024 + 512], tw, lane, -1.0f);
    #pragma unroll
    for (int k = 0; k < 8; ++k) {
        int idx = tid + 256 * k;
        int u  = idx >> 2;
        int cp = (idx & 3) << 1;
        float2 z0 = lds[(cp + 0) * 1024 + u];
        float2 z1 = lds[(cp + 1) * 1024 + u];
        *(float4*)&A[base + (size_t)u * 512 + c0 + cp] = make_float4(z0.x, z0.y, z1.x, z1.y);
    }
}

// Inverse FFT along rows -> i_low; i_high = x - i_low.
__global__ __launch_bounds__(256)
void k_row_inv(const float2* __restrict__ A, const float* __restrict__ x,
               float* __restrict__ ihigh, float* __restrict__ ilow)
{
    __shared__ float2 lds[8 * 1024];
    __shared__ float2 tw[512];
    int tid = threadIdx.x, lane = tid & 31, w = tid >> 5;
    int base_row = blockIdx.x * 8;
    build_tw(tw, tid);
    const float4* A4 = (const float4*)(A + (size_t)base_row * 512);
    #pragma unroll
    for (int k = 0; k < 8; ++k) {
        int idx = tid + 256 * k;                // 2 complex per thread-iter
        int r  = idx >> 8;
        int i0 = (idx & 255) << 1;
        float4 e = A4[idx];
        lds[r * 1024 + ((i0 + 0) & 15) * 32 + ((i0 + 0) >> 4)] = make_float2(e.x, e.y);
        lds[r * 1024 + ((i0 + 1) & 15) * 32 + ((i0 + 1) >> 4)] = make_float2(e.z, e.w);
    }
    __syncthreads();
    fft512_wave(&lds[w * 1024], &lds[w * 1024 + 512], tw, lane, -1.0f);
    const float sc = 1.0f / (512.0f * 512.0f);
    #pragma unroll
    for (int k = 0; k < 8; ++k) {
        int idx = tid + 256 * k;
        int r  = idx >> 8;
        int u0 = (idx & 255) << 1;
        size_t gid = (size_t)(base_row + r) * 512 + u0;
        float lo0 = lds[r * 1024 + u0].x * sc;
        float lo1 = lds[r * 1024 + u0 + 1].x * sc;
        float2 xv = *(const float2*)&x[gid];
        *(float2*)&ilow[gid]  = make_float2(lo0, lo1);
        *(float2*)&ihigh[gid] = make_float2(xv.x - lo0, xv.y - lo1);
    }
}

// ---------------------------------------------------------------------------
extern "C" void kernel_launch(void* const* d_in, const int* in_sizes, int n_in,
                              void* d_out, int out_size, void* d_ws, size_t ws_size,
                              hipStream_t stream)
{
    (void)in_sizes; (void)n_in; (void)out_size; (void)ws_size;
    const float* x = (const float*)d_in[0];
    float* out = (float*)d_out;

    const size_t IMG = 512 * 512;               // 262144
    float2* A     = (float2*)d_ws;                              // 48*IMG complex
    float*  gbins = (float*)((char*)d_ws + 48 * IMG * sizeof(float2));
    float*  d0ws  = gbins + 16 * 101;

    float* o_ihigh = out;                       // 48*IMG
    float* o_ilow  = out + 48 * IMG;            // 48*IMG
    float* o_d0    = out + 96 * IMG;            // 16
    float* o_mlow  = o_d0 + 16;                 // 16*IMG
    float* o_mhigh = o_mlow + 16 * IMG;         // 16*IMG

    k_zero   <<<7,    256, 0, stream>>>(gbins, 16 * 101 + 16);
    k_row_fwd<<<3072, 256, 0, stream>>>(x, A);
    k_col_fwd<<<3072, 256, 0, stream>>>(A, gbins);
    k_d0     <<<1,     32, 0, stream>>>(gbins, d0ws, o_d0);
    k_mask   <<<4096, 256, 0, stream>>>(d0ws, o_mlow, o_mhigh);
    k_col_inv<<<3072, 256, 0, stream>>>(A, d0ws);
    k_row_inv<<<3072, 256, 0, stream>>>(A, x, o_ihigh, o_ilow);
}